// EMDLoss_10625749090599
// MI455X (gfx1250) — compile-verified
//
#include <hip/hip_runtime.h>
#include <hip/hip_bf16.h>
#include <math.h>

// ---------------------------------------------------------------------------
// Sinkhorn EMD on MI455X (gfx1250).
// C[n,m] = |x_n|^2 + |y_m|^2 - 2 x.y is recomputed per sweep via
// V_WMMA_F32_16X16X4_F32 (A = 16x4 rows of x scaled by 2/eps, B = 4x16 cols
// of y, Cin = per-element additive constants), so one wmma yields the logit
// tile s = (g[m] - C[n,m])/eps directly. Stable LSE = max pass + exp-sum pass.
// Two waves cooperate per 16-row tile, each sweeping half the m-range, and
// combine max/sum via LDS -> 2048 waves per update kernel.
// ---------------------------------------------------------------------------

#define B_     8
#define N_     2048
#define EPS_   0.005f
#define ITERS_ 50
#define NT     (N_ / 16)      // 128 m-tiles
#define HT     (NT / 2)       // 64 tiles per cooperating wave

constexpr float INV_EPS     = 1.0f / EPS_;        // 200
constexpr float TWO_INV_EPS = 2.0f / EPS_;        // 400
constexpr float LOGW        = -7.62461898616f;    // -log(2048)

typedef __attribute__((ext_vector_type(2))) float v2f;
typedef __attribute__((ext_vector_type(8))) float v8f;

#define QPAD 2064   // 2048 + 16 pad -> sQ0/sQ1/sQ2 start on banks 0/16/32

// --- init: point norms + zero duals ---------------------------------------
__global__ void emd_init(const float* __restrict__ X, const float* __restrict__ Y,
                         float* __restrict__ x2, float* __restrict__ y2,
                         float* __restrict__ f, float* __restrict__ g) {
    int i = blockIdx.x * blockDim.x + threadIdx.x;
    if (i < B_ * N_) {
        float a0 = X[3*i+0], a1 = X[3*i+1], a2 = X[3*i+2];
        x2[i] = a0*a0 + a1*a1 + a2*a2;
        float b0 = Y[3*i+0], b1 = Y[3*i+1], b2 = Y[3*i+2];
        y2[i] = b0*b0 + b1*b1 + b2*b2;
        f[i] = 0.0f;
        g[i] = 0.0f;
    }
}

// --- one dual update: out[n] = eps*(logw - LSE_m((other[m]-C[n,m])/eps)) ---
// Grid: (N_/64, B_), block 256 = 8 waves = 4 pairs; pair owns a 16-row tile,
// the two waves of a pair each sweep 64 m-tiles.
__global__ void __launch_bounds__(256)
sinkhorn_update(const float* __restrict__ P, const float* __restrict__ Q,
                const float* __restrict__ p2, const float* __restrict__ q2,
                const float* __restrict__ other, float* __restrict__ outDual,
                float logw) {
    __shared__ float sQ0[QPAD];
    __shared__ float sQ1[QPAD];
    __shared__ float sQ2[QPAD];
    __shared__ float sGY[N_];
    __shared__ float sMax[4][2][16];   // [pair][half][row]
    __shared__ float sSum[4][2][16];

    const int tid   = threadIdx.x;
    const int batch = blockIdx.y;
    const int bOff  = batch * N_;

    // stage column-side data for this batch into LDS
    for (int m = tid; m < N_; m += 256) {
        int gi = bOff + m;
        sQ0[m] = Q[3*gi+0];
        sQ1[m] = Q[3*gi+1];
        sQ2[m] = Q[3*gi+2];
        sGY[m] = (other[gi] - q2[gi]) * INV_EPS;   // (g[m]-|y_m|^2)/eps
    }
    __syncthreads();

    const int lane = tid & 31;
    const int wid  = tid >> 5;
    const int pair = wid >> 1;           // 0..3 : which 16-row tile
    const int h    = wid & 1;            // 0/1  : which half of the m-range
    const int c    = lane & 15;          // column-within-tile / A-row index
    const int hi   = lane >> 4;          // 0: K=0,1 rows 0-7 | 1: K=2,3 rows 8-15
    const int n0   = blockIdx.x * 64 + pair * 16;

    // A operand (loop-invariant): rows n0+c of P, scaled by 2/eps
    const int na = bOff + n0 + c;
    v2f a;
    a[0] = (hi ? P[3*na+2] : P[3*na+0]) * TWO_INV_EPS;
    a[1] = (hi ? 0.0f      : P[3*na+1]) * TWO_INV_EPS;

    // per-D-row constants |x_row|^2/eps  (row = v + 8*hi)
    float pr[8];
#pragma unroll
    for (int v = 0; v < 8; ++v) pr[v] = p2[bOff + n0 + v + 8*hi] * INV_EPS;

    const float* qsel = hi ? sQ2 : sQ0;
    const int t0 = h * HT, t1 = t0 + HT;

    // ---- pass 1: row maxima of s = (other - C)/eps over this wave's half ----
    float mx[8];
#pragma unroll
    for (int v = 0; v < 8; ++v) mx[v] = -3.4e38f;
    for (int t = t0; t < t1; ++t) {
        int mc = 16*t + c;
        v2f b; b[0] = qsel[mc]; float q1 = sQ1[mc]; b[1] = hi ? 0.0f : q1;
        float gyc = sGY[mc];
        v8f cin;
#pragma unroll
        for (int v = 0; v < 8; ++v) cin[v] = gyc - pr[v];
        v8f d = __builtin_amdgcn_wmma_f32_16x16x4_f32(
                    false, a, false, b, (short)0, cin, false, false);
#pragma unroll
        for (int v = 0; v < 8; ++v) mx[v] = fmaxf(mx[v], d[v]);
    }
#pragma unroll
    for (int off = 1; off < 16; off <<= 1) {
#pragma unroll
        for (int v = 0; v < 8; ++v)
            mx[v] = fmaxf(mx[v], __shfl_xor(mx[v], off, 32));
    }
#pragma unroll
    for (int v = 0; v < 8; ++v)
        if (c == v) sMax[pair][h][v + 8*hi] = mx[v];
    __syncthreads();
    float mxc[8];
#pragma unroll
    for (int v = 0; v < 8; ++v) {
        int r = v + 8*hi;
        mxc[v] = fmaxf(sMax[pair][0][r], sMax[pair][1][r]);
    }

    // ---- pass 2: sum of exp(s - max) over this wave's half ----
    float sm[8];
#pragma unroll
    for (int v = 0; v < 8; ++v) sm[v] = 0.0f;
    for (int t = t0; t < t1; ++t) {
        int mc = 16*t + c;
        v2f b; b[0] = qsel[mc]; float q1 = sQ1[mc]; b[1] = hi ? 0.0f : q1;
        float gyc = sGY[mc];
        v8f cin;
#pragma unroll
        for (int v = 0; v < 8; ++v) cin[v] = gyc - pr[v];
        v8f d = __builtin_amdgcn_wmma_f32_16x16x4_f32(
                    false, a, false, b, (short)0, cin, false, false);
#pragma unroll
        for (int v = 0; v < 8; ++v) sm[v] += __expf(d[v] - mxc[v]);
    }
#pragma unroll
    for (int off = 1; off < 16; off <<= 1) {
#pragma unroll
        for (int v = 0; v < 8; ++v)
            sm[v] += __shfl_xor(sm[v], off, 32);
    }
#pragma unroll
    for (int v = 0; v < 8; ++v)
        if (c == v) sSum[pair][h][v + 8*hi] = sm[v];
    __syncthreads();

    // half 0 writes out[row] = eps*(logw - (max + log(sum0+sum1)))
    if (h == 0) {
#pragma unroll
        for (int v = 0; v < 8; ++v) {
            int r = v + 8*hi;
            float smc = sSum[pair][0][r] + sSum[pair][1][r];
            float val = EPS_ * (logw - (mxc[v] + __logf(smc)));
            if (c == v) outDual[bOff + n0 + r] = val;
        }
    }
}

// --- loss: sum_{b,n,m} exp((f+g-C)/eps) * max(C,0), per-block partials ----
// Same pair/half decomposition; the block-wide sum already merges halves.
__global__ void __launch_bounds__(256)
emd_loss(const float* __restrict__ X, const float* __restrict__ Y,
         const float* __restrict__ x2, const float* __restrict__ y2,
         const float* __restrict__ f, const float* __restrict__ g,
         float* __restrict__ partials) {
    __shared__ float sQ0[QPAD];
    __shared__ float sQ1[QPAD];
    __shared__ float sQ2[QPAD];
    __shared__ float sGY[N_];
    __shared__ float sG[N_];
    __shared__ float waveSum[8];

    const int tid   = threadIdx.x;
    const int batch = blockIdx.y;
    const int bOff  = batch * N_;

    for (int m = tid; m < N_; m += 256) {
        int gi = bOff + m;
        sQ0[m] = Y[3*gi+0];
        sQ1[m] = Y[3*gi+1];
        sQ2[m] = Y[3*gi+2];
        float gv = g[gi];
        sG[m]  = gv;
        sGY[m] = (gv - y2[gi]) * INV_EPS;
    }
    __syncthreads();

    const int lane = tid & 31;
    const int wid  = tid >> 5;
    const int pair = wid >> 1;
    const int h    = wid & 1;
    const int c    = lane & 15;
    const int hi   = lane >> 4;
    const int n0   = blockIdx.x * 64 + pair * 16;

    const int na = bOff + n0 + c;
    v2f a;
    a[0] = (hi ? X[3*na+2] : X[3*na+0]) * TWO_INV_EPS;
    a[1] = (hi ? 0.0f      : X[3*na+1]) * TWO_INV_EPS;

    float fr[8], fp[8];
#pragma unroll
    for (int v = 0; v < 8; ++v) {
        int r = bOff + n0 + v + 8*hi;
        fr[v] = f[r];
        fp[v] = (fr[v] - x2[r]) * INV_EPS;   // (f[n]-|x_n|^2)/eps
    }

    const float* qsel = hi ? sQ2 : sQ0;
    const int t0 = h * HT, t1 = t0 + HT;

    float acc = 0.0f;
    for (int t = t0; t < t1; ++t) {
        int mc = 16*t + c;
        v2f b; b[0] = qsel[mc]; float q1 = sQ1[mc]; b[1] = hi ? 0.0f : q1;
        float gyc = sGY[mc];
        float gc  = sG[mc];
        v8f cin;
#pragma unroll
        for (int v = 0; v < 8; ++v) cin[v] = gyc + fp[v];
        // d = (f[n] + g[m] - C[n,m]) / eps
        v8f d = __builtin_amdgcn_wmma_f32_16x16x4_f32(
                    false, a, false, b, (short)0, cin, false, false);
#pragma unroll
        for (int v = 0; v < 8; ++v) {
            float e  = __expf(d[v]);                       // P element
            float Cv = fr[v] + gc - EPS_ * d[v];           // recover C
            Cv = fmaxf(Cv, 0.0f);
            acc += e * Cv;
        }
    }
    // reduce across the full wave (order-deterministic)
#pragma unroll
    for (int off = 1; off < 32; off <<= 1)
        acc += __shfl_xor(acc, off, 32);
    if (lane == 0) waveSum[wid] = acc;
    __syncthreads();
    if (tid == 0) {
        float s = 0.0f;
#pragma unroll
        for (int w = 0; w < 8; ++w) s += waveSum[w];
        partials[blockIdx.y * gridDim.x + blockIdx.x] = s;
    }
}

__global__ void emd_reduce(const float* __restrict__ partials, float* __restrict__ out,
                           int nPart) {
    __shared__ float s[256];
    float v = 0.0f;
    for (int i = threadIdx.x; i < nPart; i += 256) v += partials[i];
    s[threadIdx.x] = v;
    __syncthreads();
    for (int st = 128; st > 0; st >>= 1) {
        if (threadIdx.x < st) s[threadIdx.x] += s[threadIdx.x + st];
        __syncthreads();
    }
    if (threadIdx.x == 0) out[0] = s[0] * (1.0f / (float)B_);
}

// ---------------------------------------------------------------------------
extern "C" void kernel_launch(void* const* d_in, const int* in_sizes, int n_in,
                              void* d_out, int out_size, void* d_ws, size_t ws_size,
                              hipStream_t stream) {
    const float* x = (const float*)d_in[0];   // [B,N,3]
    const float* y = (const float*)d_in[1];   // [B,N,3]
    float* out = (float*)d_out;

    float* w        = (float*)d_ws;
    float* f        = w;                  // [B*N]
    float* g        = w + 1 * B_ * N_;    // [B*N]
    float* x2       = w + 2 * B_ * N_;    // [B*N]
    float* y2       = w + 3 * B_ * N_;    // [B*N]
    float* partials = w + 4 * B_ * N_;    // [256]

    emd_init<<<(B_ * N_ + 255) / 256, 256, 0, stream>>>(x, y, x2, y2, f, g);

    dim3 grid(N_ / 64, B_);   // 32 x 8 blocks, 4 row-tiles each, 2 waves/tile
    for (int it = 0; it < ITERS_; ++it) {
        // f update uses current g; g update uses the fresh f (matches scan body)
        sinkhorn_update<<<grid, 256, 0, stream>>>(x, y, x2, y2, g, f, LOGW);
        sinkhorn_update<<<grid, 256, 0, stream>>>(y, x, y2, x2, f, g, LOGW);
    }

    emd_loss<<<grid, 256, 0, stream>>>(x, y, x2, y2, f, g, partials);
    emd_reduce<<<1, 256, 0, stream>>>(partials, out, grid.x * grid.y);
}